// ForwardBPTTCell_58299886076334
// MI455X (gfx1250) — compile-verified
//
#include <hip/hip_runtime.h>
#include <math.h>

#define H 256
#define D 128
#define O 128
#define T 1024
#define LDA 257   // padded LDS row stride: 257 % 64 == 1 -> conflict-free column access

typedef __attribute__((ext_vector_type(2))) float v2f;
typedef __attribute__((ext_vector_type(4))) float v4f;
typedef __attribute__((ext_vector_type(8))) float v8f;

// ---- optional gfx1250 async global->LDS path (guarded; falls back cleanly) ----
#if defined(__has_builtin)
#if __has_builtin(__builtin_amdgcn_global_load_async_to_lds_b32) && \
    __has_builtin(__builtin_amdgcn_s_wait_asynccnt)
#define HAVE_ASYNC_LDS 1
#endif
#endif

#ifdef HAVE_ASYNC_LDS
typedef __attribute__((address_space(1))) int gas_int;   // global (AS1)
typedef __attribute__((address_space(3))) int las_int;   // LDS    (AS3)

__device__ __forceinline__ void async_g2l_b32(const float* g, float* l) {
  __builtin_amdgcn_global_load_async_to_lds_b32((gas_int*)g, (las_int*)l, 0, 0);
}
#endif

// Stage a 256x256 f32 tile from global into LDS with 257-padded rows.
__device__ __forceinline__ void stage_tile(const float* __restrict__ G,
                                           float* __restrict__ Lds, int tid) {
#ifdef HAVE_ASYNC_LDS
  for (int idx = tid; idx < H * H; idx += 256)
    async_g2l_b32(G + idx, Lds + ((idx >> 8) * LDA + (idx & 255)));
  __builtin_amdgcn_s_wait_asynccnt(0);
#else
  for (int idx = tid; idx < H * H; idx += 256)
    Lds[(idx >> 8) * LDA + (idx & 255)] = G[idx];
#endif
}

// ---------------------------------------------------------------------------
// WMMA f32 GEMM:  C = (A [- A2]) x (B | B^T) [+ bias],  dims %16, K %8.
// One 16x16 C tile per wave, V_WMMA_F32_16X16X4_F32, K unrolled x2.
// A 16x4 : lane&15 = M row; VGPR0/1 = K + (lane>>4)*2 + {0,1}
// B 4x16 : lane&15 = N col; VGPR0/1 = K + (lane>>4)*2 + {0,1}
// C 16x16: VGPR e -> M = e + (lane>>4)*8, N = lane&15
// ---------------------------------------------------------------------------
template <bool SUB, bool TRANSB, bool BIAS>
__global__ __launch_bounds__(256)
void wmma_gemm_f32(const float* __restrict__ A,
                   const float* __restrict__ A2,
                   const float* __restrict__ B,
                   const float* __restrict__ bias,
                   float* __restrict__ C,
                   int M, int N, int K, int lda, int ldb, int ldc)
{
  const int lane = threadIdx.x & 31;
  const int wave = threadIdx.x >> 5;
  const int wavesPerBlock = blockDim.x >> 5;
  const int nTilesN = N >> 4;
  const int nTiles  = (M >> 4) * nTilesN;
  const int mn = lane & 15;          // M row for A, N col for B/C
  const int kk = (lane >> 4) << 1;   // 0 or 2

  for (int tile = blockIdx.x * wavesPerBlock + wave; tile < nTiles;
       tile += gridDim.x * wavesPerBlock) {
    const int m0 = (tile / nTilesN) << 4;
    const int n0 = (tile % nTilesN) << 4;
    const float* arow = A + (size_t)(m0 + mn) * lda + kk;
    const float* arow2 = SUB ? (A2 + (size_t)(m0 + mn) * lda + kk) : nullptr;
    v8f acc = {};
    for (int k0 = 0; k0 < K; k0 += 8) {
      v2f a0, a1, b0, b1;
      a0.x = arow[k0];     a0.y = arow[k0 + 1];
      a1.x = arow[k0 + 4]; a1.y = arow[k0 + 5];
      if (SUB) {
        a0.x -= arow2[k0];     a0.y -= arow2[k0 + 1];
        a1.x -= arow2[k0 + 4]; a1.y -= arow2[k0 + 5];
      }
      if (TRANSB) {
        const float* bp = B + (size_t)(n0 + mn) * ldb + kk;
        b0.x = bp[k0];     b0.y = bp[k0 + 1];
        b1.x = bp[k0 + 4]; b1.y = bp[k0 + 5];
      } else {
        b0.x = B[(size_t)(k0 + kk)     * ldb + (n0 + mn)];
        b0.y = B[(size_t)(k0 + kk + 1) * ldb + (n0 + mn)];
        b1.x = B[(size_t)(k0 + 4 + kk)     * ldb + (n0 + mn)];
        b1.y = B[(size_t)(k0 + 4 + kk + 1) * ldb + (n0 + mn)];
      }
      acc = __builtin_amdgcn_wmma_f32_16x16x4_f32(false, a0, false, b0,
                                                  (short)0, acc, false, false);
      acc = __builtin_amdgcn_wmma_f32_16x16x4_f32(false, a1, false, b1,
                                                  (short)0, acc, false, false);
    }
    const float bb = BIAS ? bias[n0 + mn] : 0.0f;
    const int rbase = (lane >> 4) << 3;   // 0 or 8
    #pragma unroll
    for (int e = 0; e < 8; ++e)
      C[(size_t)(m0 + rbase + e) * ldc + (n0 + mn)] = acc[e] + bb;
  }
}

// ---------------------------------------------------------------------------
__global__ __launch_bounds__(256)
void transpose256(const float* __restrict__ src, float* __restrict__ dst)
{
  int idx = blockIdx.x * 256 + threadIdx.x;
  int r = idx >> 8, c = idx & 255;
  dst[c * 256 + r] = src[idx];
}

// ---------------------------------------------------------------------------
// Sequential GRU recurrence (1 workgroup, thread i owns hidden unit i).
// ---------------------------------------------------------------------------
__global__ __launch_bounds__(256)
void gru_recurrence(const float* __restrict__ gx,     // T x 3H  (z|r|n preacts)
                    const float* __restrict__ WhzT,
                    const float* __restrict__ WhrT,
                    const float* __restrict__ WhnT,
                    const float* __restrict__ b_hn,
                    float* __restrict__ hseq,
                    float* __restrict__ c1, float* __restrict__ c2,
                    float* __restrict__ c3, float* __restrict__ zz)
{
  __shared__ __align__(16) float hs[H];
  const int i = threadIdx.x;
  hs[i] = 0.0f;
  const v4f* wz = (const v4f*)(WhzT + (size_t)i * H);
  const v4f* wr = (const v4f*)(WhrT + (size_t)i * H);
  const v4f* wn = (const v4f*)(WhnT + (size_t)i * H);
  const float bhn = b_hn[i];
  __syncthreads();

  for (int t = 0; t < T; ++t) {
    float az = 0.f, ar = 0.f, an = 0.f;
    #pragma unroll 4
    for (int j = 0; j < H / 4; ++j) {
      v4f h4 = *(const v4f*)(hs + 4 * j);
      v4f z4 = wz[j], r4 = wr[j], n4 = wn[j];
      az += h4.x * z4.x + h4.y * z4.y + h4.z * z4.z + h4.w * z4.w;
      ar += h4.x * r4.x + h4.y * r4.y + h4.z * r4.z + h4.w * r4.w;
      an += h4.x * n4.x + h4.y * n4.y + h4.z * n4.z + h4.w * n4.w;
    }
    const float* g = gx + (size_t)t * (3 * H);
    float z = 1.f / (1.f + expf(-(g[i] + az)));
    float r = 1.f / (1.f + expf(-(g[H + i] + ar)));
    float a = an + bhn;
    float n = tanhf(g[2 * H + i] + r * a);
    float hold = hs[i];
    float hnew = (1.f - z) * n + z * hold;
    float npr = 1.f - n * n;
    size_t o = (size_t)t * H + i;
    hseq[o] = hnew;
    c1[o] = z * (1.f - z) * (hold - n);
    c2[o] = (1.f - z) * npr * r;
    c3[o] = (1.f - z) * npr * a * r * (1.f - r);
    zz[o] = z;
    __syncthreads();
    hs[i] = hnew;
    __syncthreads();
  }
}

// ---------------------------------------------------------------------------
// Jt[t][j][i] = Whz[j][i]*c1 + Whn[j][i]*c2 + Whr[j][i]*c3 + (i==j)*z
// ---------------------------------------------------------------------------
__global__ __launch_bounds__(256)
void build_jac(const float* __restrict__ Whz, const float* __restrict__ Whn,
               const float* __restrict__ Whr,
               const float* __restrict__ c1, const float* __restrict__ c2,
               const float* __restrict__ c3, const float* __restrict__ zz,
               float* __restrict__ jac)
{
  const int t  = blockIdx.y;
  const int j  = (blockIdx.x << 2) + (threadIdx.x >> 6);
  const int i0 = (threadIdx.x & 63) << 2;
  const v4f wz = *(const v4f*)(Whz + (size_t)j * H + i0);
  const v4f wn = *(const v4f*)(Whn + (size_t)j * H + i0);
  const v4f wr = *(const v4f*)(Whr + (size_t)j * H + i0);
  const v4f a1 = *(const v4f*)(c1 + (size_t)t * H + i0);
  const v4f a2 = *(const v4f*)(c2 + (size_t)t * H + i0);
  const v4f a3 = *(const v4f*)(c3 + (size_t)t * H + i0);
  const v4f az = *(const v4f*)(zz + (size_t)t * H + i0);
  v4f o;
  o.x = wz.x * a1.x + wn.x * a2.x + wr.x * a3.x + ((i0 + 0) == j ? az.x : 0.f);
  o.y = wz.y * a1.y + wn.y * a2.y + wr.y * a3.y + ((i0 + 1) == j ? az.y : 0.f);
  o.z = wz.z * a1.z + wn.z * a2.z + wr.z * a3.z + ((i0 + 2) == j ? az.z : 0.f);
  o.w = wz.w * a1.w + wn.w * a2.w + wr.w * a3.w + ((i0 + 3) == j ? az.w : 0.f);
  *(v4f*)(jac + ((size_t)t * H + j) * H + i0) = o;
}

// ---------------------------------------------------------------------------
// Batched in-place LU with partial pivoting. One workgroup (256 thr) per t.
// Whole 256x256 matrix in LDS (CDNA5: 320 KB/WGP). Pivot argmax via
// wave32 __shfl_xor reduction (ds_permute, no barriers) + tiny combine.
// ---------------------------------------------------------------------------
__global__ __launch_bounds__(256)
void batched_lu(float* __restrict__ jac, int* __restrict__ piv)
{
  extern __shared__ float lds[];
  float* Am   = lds;                 // 256*257
  float* redv = lds + 256 * LDA;     // 8
  int*   redi = (int*)(redv + 8);    // 8
  const int tid  = threadIdx.x;
  const int lane = tid & 31;
  const int wv   = tid >> 5;
  float* G = jac + (size_t)blockIdx.x * (H * H);

  stage_tile(G, Am, tid);
  __syncthreads();

  for (int k = 0; k < H; ++k) {
    // --- pivot search over rows k..255 of column k
    int r = k + tid;
    float v = -1.f; int pi = k;
    if (r < H) { v = fabsf(Am[r * LDA + k]); pi = r; }
    #pragma unroll
    for (int s = 16; s > 0; s >>= 1) {
      float ov = __shfl_xor(v, s, 32);
      int   oi = __shfl_xor(pi, s, 32);
      if (ov > v) { v = ov; pi = oi; }
    }
    if (lane == 0) { redv[wv] = v; redi[wv] = pi; }
    __syncthreads();
    if (tid == 0) {
      float bv = redv[0]; int bp = redi[0];
      #pragma unroll
      for (int w = 1; w < 8; ++w)
        if (redv[w] > bv) { bv = redv[w]; bp = redi[w]; }
      redi[0] = bp;
      piv[blockIdx.x * H + k] = bp;
    }
    __syncthreads();
    const int p = redi[0];
    if (p != k) {  // swap full rows k<->p, thread = column
      float t1 = Am[k * LDA + tid];
      Am[k * LDA + tid] = Am[p * LDA + tid];
      Am[p * LDA + tid] = t1;
    }
    __syncthreads();
    // --- eliminate: thread owns row k+1+tid
    const float inv = 1.f / Am[k * LDA + k];
    const int r2 = k + 1 + tid;
    if (r2 < H) {
      float l = Am[r2 * LDA + k] * inv;
      Am[r2 * LDA + k] = l;
      const float* pk = Am + k * LDA;
      float* pr = Am + r2 * LDA;
      for (int j = k + 1; j < H; ++j) pr[j] -= l * pk[j];
    }
    __syncthreads();
  }

  for (int idx = tid; idx < H * H; idx += 256)
    G[idx] = Am[(idx >> 8) * LDA + (idx & 255)];
}

// ---------------------------------------------------------------------------
// Sequential delta chain: delta[t] = Jt[t]^-1 (delta[t-1] - inst[t-1]).
// LU(t) staged by all 8 waves; the triangular solves run in a SINGLE wave32
// (lane owns rows lane+32m): intra-wave LDS ops are in-order, so only free
// wave_barriers are needed per k instead of s_barrier.
// ---------------------------------------------------------------------------
__global__ __launch_bounds__(256)
void delta_chain(const float* __restrict__ lu, const int* __restrict__ piv,
                 const float* __restrict__ inst, float* __restrict__ delta)
{
  extern __shared__ float lds[];
  float* LU  = lds;                  // 256*257
  float* rhs = lds + 256 * LDA;      // 256
  int*   pv  = (int*)(rhs + 256);    // 256
  const int tid = threadIdx.x;
  float dprev = 0.f;
  delta[tid] = 0.f;                  // delta_seq[0] = 0 (zero rhs -> zero solve)

  for (int t = 1; t < T; ++t) {
    const float* G = lu + (size_t)t * (H * H);
    stage_tile(G, LU, tid);
    rhs[tid] = dprev - inst[(size_t)(t - 1) * H + tid];
    pv[tid]  = piv[t * H + tid];
    __syncthreads();

    if (tid < 32) {
      const int lane = tid;
      if (lane == 0) {               // apply pivot swaps in order
        for (int k = 0; k < H; ++k) {
          int p = pv[k];
          float tmp = rhs[k]; rhs[k] = rhs[p]; rhs[p] = tmp;
        }
      }
      __builtin_amdgcn_wave_barrier();
      // forward substitution, unit-lower
      for (int k = 0; k < H - 1; ++k) {
        float xk = rhs[k];
        #pragma unroll
        for (int m = 0; m < 8; ++m) {
          int i = lane + 32 * m;
          if (i > k) rhs[i] -= LU[i * LDA + k] * xk;
        }
        __builtin_amdgcn_wave_barrier();
      }
      // backward substitution
      for (int k = H - 1; k >= 0; --k) {
        if (lane == (k & 31)) rhs[k] /= LU[k * LDA + k];
        __builtin_amdgcn_wave_barrier();
        float xk = rhs[k];
        #pragma unroll
        for (int m = 0; m < 8; ++m) {
          int i = lane + 32 * m;
          if (i < k) rhs[i] -= LU[i * LDA + k] * xk;
        }
        __builtin_amdgcn_wave_barrier();
      }
    }
    __syncthreads();

    dprev = rhs[tid];
    delta[(size_t)t * H + tid] = dprev;
    if (t + 1 < T)   // warm L2 for the next LU factor
      __builtin_prefetch(lu + (size_t)(t + 1) * (H * H) + tid * 256, 0, 1);
    __syncthreads();
  }
}

// ---------------------------------------------------------------------------
extern "C" void kernel_launch(void* const* d_in, const int* in_sizes, int n_in,
                              void* d_out, int out_size, void* d_ws, size_t ws_size,
                              hipStream_t stream)
{
  (void)in_sizes; (void)n_in; (void)out_size; (void)ws_size;
  const float* x_seq = (const float*)d_in[0];
  const float* y_seq = (const float*)d_in[1];
  const float* Wiz   = (const float*)d_in[2];
  const float* Whz   = (const float*)d_in[3];
  const float* b_z   = (const float*)d_in[4];
  const float* Wir   = (const float*)d_in[5];
  const float* Whr   = (const float*)d_in[6];
  const float* b_r   = (const float*)d_in[7];
  const float* Win   = (const float*)d_in[8];
  const float* Whn   = (const float*)d_in[9];
  const float* b_n   = (const float*)d_in[10];
  const float* b_hn  = (const float*)d_in[11];
  const float* Wo    = (const float*)d_in[12];
  const float* b_o   = (const float*)d_in[13];

  float* out_seq   = (float*)d_out;                 // T*O
  float* delta_seq = out_seq + (size_t)T * O;       // T*H
  float* inst_seq  = delta_seq + (size_t)T * H;     // T*H
  float* jac_seq   = inst_seq + (size_t)T * H;      // T*H*H

  float* ws   = (float*)d_ws;                       // ~10 MB total
  float* gx   = ws;                                 // T*3H
  float* hseq = gx + (size_t)T * 3 * H;             // T*H
  float* c1   = hseq + (size_t)T * H;
  float* c2   = c1 + (size_t)T * H;
  float* c3   = c2 + (size_t)T * H;
  float* zz   = c3 + (size_t)T * H;
  float* WhzT = zz + (size_t)T * H;
  float* WhrT = WhzT + H * H;
  float* WhnT = WhrT + H * H;
  int*   piv  = (int*)(WhnT + H * H);               // T*H ints

  // 1. transpose hidden weights for per-thread-contiguous column access
  transpose256<<<256, 256, 0, stream>>>(Whz, WhzT);
  transpose256<<<256, 256, 0, stream>>>(Whr, WhrT);
  transpose256<<<256, 256, 0, stream>>>(Whn, WhnT);

  // 2. batched input-gate preactivations: gx = x_seq @ W + b (WMMA f32)
  wmma_gemm_f32<false,false,true><<<128, 256, 0, stream>>>(
      x_seq, nullptr, Wiz, b_z, gx + 0,     T, H, D, D, H, 3 * H);
  wmma_gemm_f32<false,false,true><<<128, 256, 0, stream>>>(
      x_seq, nullptr, Wir, b_r, gx + H,     T, H, D, D, H, 3 * H);
  wmma_gemm_f32<false,false,true><<<128, 256, 0, stream>>>(
      x_seq, nullptr, Win, b_n, gx + 2 * H, T, H, D, D, H, 3 * H);

  // 3. sequential GRU recurrence -> h_seq + Jt scale vectors
  gru_recurrence<<<1, 256, 0, stream>>>(gx, WhzT, WhrT, WhnT, b_hn,
                                        hseq, c1, c2, c3, zz);

  // 4. out_seq = h_seq @ Wo + b_o   (WMMA f32)
  wmma_gemm_f32<false,false,true><<<64, 256, 0, stream>>>(
      hseq, nullptr, Wo, b_o, out_seq, T, O, H, H, O, O);

  // 5. inst_seq = (out_seq - y_seq) @ Wo^T   (WMMA f32, fused diff + transB)
  wmma_gemm_f32<true,true,false><<<128, 256, 0, stream>>>(
      out_seq, y_seq, Wo, nullptr, inst_seq, T, H, O, O, O, H);

  // 6. materialize jac_seq (268 MB, bandwidth-bound b128 stores)
  dim3 bj(64, T);
  build_jac<<<bj, 256, 0, stream>>>(Whz, Whn, Whr, c1, c2, c3, zz, jac_seq);

  // 7. LU-factorize all 1024 Jt's in parallel, in place (256KB LDS tile/WG)
  size_t luLds = (size_t)(256 * LDA + 8) * sizeof(float) + 8 * sizeof(int);
  batched_lu<<<T, 256, luLds, stream>>>(jac_seq, piv);

  // 8. sequential chain of O(H^2) triangular solves (single-wave solver)
  size_t slvLds = (size_t)(256 * LDA + 256) * sizeof(float) + 256 * sizeof(int);
  delta_chain<<<1, 256, slvLds, stream>>>(jac_seq, piv, inst_seq, delta_seq);

  // 9. rebuild jac_seq output (LU destroyed it; rebuild is cheap, same kernel)
  build_jac<<<bj, 256, 0, stream>>>(Whz, Whn, Whr, c1, c2, c3, zz, jac_seq);
}